// net_46411416600704
// MI455X (gfx1250) — compile-verified
//
#include <hip/hip_runtime.h>
#include <stdint.h>

typedef float v2f __attribute__((ext_vector_type(2)));
typedef float v8f __attribute__((ext_vector_type(8)));

#define LDA_P 36   // padded LDS stride (dwords) for A tile rows; 16-B aligned, spreads banks
#define KT 32      // K chunk per LDS stage

// Async global->LDS copy (CDNA5), tracked by ASYNCcnt.
__device__ __forceinline__ void async_cp_b128(uint32_t lds, const float* g) {
  asm volatile("global_load_async_to_lds_b128 %0, %1, off" :: "v"(lds), "v"(g) : "memory");
}
template<int CNT>
__device__ __forceinline__ void wait_asynccnt() {
  asm volatile("s_wait_asynccnt %0" :: "n"(CNT) : "memory");
}

constexpr int ilog2c(int v) { int s = 0; while ((1 << s) < v) ++s; return s; }

// Row-gathered WMMA f32 GEMM, double-buffered async LDS staging.
//   C[row, colofs+n] = epi( alpha * sum_k A[g(row), k] * B[k, n] )
// Block: 256 threads (8 wave32), 64 x N tile; N in {32,64,128}; K % 32 == 0, K >= 2*KT.
template<int N>
__global__ __launch_bounds__(256) void gemm_f32_wmma(
    const float* __restrict__ A, const int* __restrict__ rowidx, int lda,
    const float* __restrict__ B, float* __restrict__ C, int ldc, int colofs,
    int K, const float* __restrict__ alpha_ptr, float alpha_sign,
    const float* __restrict__ addsrc, const float* __restrict__ bias, int relu)
{
  constexpr int NT    = N / 32;        // wmma tiles per wave (compile-time)
  constexpr int LDBP  = N + 4;
  constexpr int ASZ   = 64 * LDA_P;    // floats per A buffer
  constexpr int BSZ   = KT * LDBP;     // floats per B buffer
  constexpr int NQ    = N / 4;         // float4 per B row
  constexpr int NQSH  = ilog2c(NQ);
  constexpr int BRS   = 256 >> NQSH;   // B rows covered per 256-thread pass
  constexpr int NBJ   = (8 * N) / 256; // async b128 per thread for B (= N/32)
  constexpr int PERCHUNK = 2 + NBJ;    // async insts in flight for the next chunk

  __shared__ float lA[2 * ASZ];
  __shared__ float lB[2 * BSZ];

  const int tid  = threadIdx.x;
  const int wave = tid >> 5;
  const int lane = tid & 31;
  const int half = lane >> 4;
  const int l16  = lane & 15;
  const int mi      = wave & 3;        // 4 M-subtiles of 16 rows
  const int njstart = wave >> 2;       // 2-way N split across waves
  const int m0 = blockIdx.x * 64;

  const float alpha = alpha_sign * (alpha_ptr ? alpha_ptr[0] : 1.0f);

  // ---- per-thread staging geometry, hoisted out of the K loop ----
  const int arow = tid >> 3;                 // A rows: arow and arow+32
  const int acs  = (tid & 7) * 4;
  const int g0 = rowidx ? rowidx[m0 + arow]      : (m0 + arow);
  const int g1 = rowidx ? rowidx[m0 + arow + 32] : (m0 + arow + 32);
  const float* Ar0 = A + (size_t)g0 * lda + acs;
  const float* Ar1 = A + (size_t)g1 * lda + acs;
  const uint32_t ldsA0 = (uint32_t)(size_t)&lA[arow * LDA_P + acs];
  const uint32_t ldsA1 = (uint32_t)(size_t)&lA[(arow + 32) * LDA_P + acs];

  const int brow = tid >> NQSH;
  const int bc4  = (tid & (NQ - 1)) * 4;
  const float* Bth = B + (size_t)brow * N + bc4;
  uint32_t ldsBj[NBJ];
#pragma unroll
  for (int j = 0; j < NBJ; ++j)
    ldsBj[j] = (uint32_t)(size_t)&lB[(brow + j * BRS) * LDBP + bc4];

  auto stage = [&](int k0, int buf) {
    const uint32_t ao = buf ? (uint32_t)(ASZ * 4) : 0u;
    const uint32_t bo = buf ? (uint32_t)(BSZ * 4) : 0u;
    async_cp_b128(ldsA0 + ao, Ar0 + k0);
    async_cp_b128(ldsA1 + ao, Ar1 + k0);
#pragma unroll
    for (int j = 0; j < NBJ; ++j)
      async_cp_b128(ldsBj[j] + bo, Bth + (size_t)(k0 + j * BRS) * N);
  };

  v8f acc[NT] = {};

  auto compute = [&](int buf) {
    const float* sA = lA + buf * ASZ;
    const float* sB = lB + buf * BSZ;
#pragma unroll
    for (int kk = 0; kk < KT; kk += 4) {
      // A frag per ISA 16x4 f32 layout: VGPR0 = K {0|2}, VGPR1 = K {1|3} by half-wave.
      v2f a = *(const v2f*)&sA[(mi * 16 + l16) * LDA_P + kk + 2 * half];
#pragma unroll
      for (int t = 0; t < NT; ++t) {
        const int c = (njstart + 2 * t) * 16 + l16;
        v2f b;
        b.x = sB[(kk + half) * LDBP + c];       // B 4x16: VGPR0 = K {0|1}
        b.y = sB[(kk + 2 + half) * LDBP + c];   //          VGPR1 = K {2|3}
        acc[t] = __builtin_amdgcn_wmma_f32_16x16x4_f32(
            false, a, false, b, (short)0, acc[t], false, false);
      }
    }
  };

  // Double-buffered pipeline, last chunk peeled: steady-state body is branch-free.
  stage(0, 0);
  int buf = 0;
  int k0 = 0;
  for (; k0 + KT < K; k0 += KT) {
    stage(k0 + KT, buf ^ 1);     // prefetch next chunk into other buffer
    wait_asynccnt<PERCHUNK>();   // in-order ASYNCcnt: current chunk has landed
    __syncthreads();
    compute(buf);
    __syncthreads();             // everyone done reading buf^1 from prior iter
    buf ^= 1;
  }
  wait_asynccnt<0>();
  __syncthreads();
  compute(buf);                  // epilogue below is register-only; no barrier needed

  // C/D 16x16 f32 layout: VGPR i -> M = i (lanes 0-15) / i+8 (lanes 16-31), N = lane%16.
#pragma unroll
  for (int t = 0; t < NT; ++t) {
    const int col = colofs + (njstart + 2 * t) * 16 + l16;
#pragma unroll
    for (int i = 0; i < 8; ++i) {
      const int row = m0 + mi * 16 + 8 * half + i;
      float v = alpha * acc[t][i];
      if (addsrc) v += addsrc[(size_t)row * ldc + col];
      if (bias)   v += bias[col];
      if (relu)   v = fmaxf(v, 0.0f);
      C[(size_t)row * ldc + col] = v;
    }
  }
}

// u[n_id[j], :] += src[(gathered ? n_id[j] : j), :]  — densifies the column gather.
__global__ __launch_bounds__(256) void scatter_kernel(
    const float* __restrict__ src, const int* __restrict__ idx,
    float* __restrict__ dst, int rows, int cols, int src_is_gathered)
{
  int g = blockIdx.x * blockDim.x + threadIdx.x;
  if (g >= rows * cols) return;
  int j = g / cols, d = g - j * cols;
  int r = idx[j];
  int sr = src_is_gathered ? r : j;
  atomicAdd(dst + (size_t)r * cols + d, src[(size_t)sr * cols + d]);
}

__global__ __launch_bounds__(256) void copy_h1_to_z(
    const float* __restrict__ h1, float* __restrict__ z)
{
  int g = blockIdx.x * blockDim.x + threadIdx.x;
  if (g >= 2048 * 64) return;
  int i = g >> 6, d = g & 63;
  z[(size_t)i * 192 + d] = h1[(size_t)i * 64 + d];
}

// Mean over 16 contiguous edges per target (dst = repeat(arange(1024),16)),
// and build conv input azw = [ z[:1024] | agg ] (1024 x 384).
__global__ __launch_bounds__(256) void agg_build_azw(
    const float* __restrict__ z, const int* __restrict__ edge_index,
    float* __restrict__ azw)
{
  int g = blockIdx.x * blockDim.x + threadIdx.x;
  if (g >= 1024 * 192) return;
  int i = g / 192, d = g - i * 192;
  const int* src = edge_index;            // edge_index[0] = src, first E=16384 ints
  float s = 0.0f;
#pragma unroll
  for (int e = 0; e < 16; ++e)
    s += z[(size_t)src[i * 16 + e] * 192 + d];
  azw[(size_t)i * 384 + 192 + d] = s * (1.0f / 16.0f);
  azw[(size_t)i * 384 + d]       = z[(size_t)i * 192 + d];
}

// Wcat = [W_self; W_neigh]  (384 x 128)
__global__ __launch_bounds__(256) void build_wcat(
    const float* __restrict__ ws_, const float* __restrict__ wn,
    float* __restrict__ wcat)
{
  int g = blockIdx.x * blockDim.x + threadIdx.x;
  if (g >= 384 * 128) return;
  int r = g >> 7, c = g & 127;
  wcat[g] = (r < 192) ? ws_[r * 128 + c] : wn[(r - 192) * 128 + c];
}

extern "C" void kernel_launch(void* const* d_in, const int* in_sizes, int n_in,
                              void* d_out, int out_size, void* d_ws, size_t ws_size,
                              hipStream_t stream) {
  (void)in_sizes; (void)n_in; (void)out_size; (void)ws_size;
  const float* x       = (const float*)d_in[0];
  const float* tau     = (const float*)d_in[1];
  const float* L       = (const float*)d_in[2];
  const float* K0      = (const float*)d_in[3];
  const float* K1      = (const float*)d_in[4];
  const float* W_self  = (const float*)d_in[5];
  const float* W_neigh = (const float*)d_in[6];
  const float* b_conv  = (const float*)d_in[7];
  const float* W1      = (const float*)d_in[8];
  const float* b1      = (const float*)d_in[9];
  const float* W2      = (const float*)d_in[10];
  const float* b2      = (const float*)d_in[11];
  const int*   n_id    = (const int*)d_in[12];
  const int*   edge_ix = (const int*)d_in[13];
  float* out = (float*)d_out;

  float* ws  = (float*)d_ws;
  float* y   = ws;                    // 8192*32
  float* u   = y   + 8192 * 32;       // 8192*32
  float* h1  = u   + 8192 * 32;       // 4096*64
  float* u2  = h1  + 4096 * 64;       // 8192*64
  float* z   = u2  + 8192 * 64;       // 2048*192
  float* azw = z   + 2048 * 192;      // 1024*384
  float* wc  = azw + 1024 * 384;      // 384*128
  float* hc  = wc  + 384 * 128;       // 1024*128
  float* hm  = hc  + 1024 * 128;      // 1024*64

  dim3 blk(256);

  // 1) diffusion: y = x - tau*(L@x)         M=8192 N=32 K=8192
  gemm_f32_wmma<32><<<8192 / 64, blk, 0, stream>>>(L, nullptr, 8192, x, y, 32, 0,
                                                   8192, tau, -1.0f, x, nullptr, 0);
  // 2) densify gather: u[p] = count_p * y[p]
  hipMemsetAsync(u, 0, (size_t)8192 * 32 * 4, stream);
  scatter_kernel<<<(6144 * 32 + 255) / 256, blk, 0, stream>>>(y, n_id, u, 6144, 32, 1);
  // 3) hop1: h1 = [K0[n_id[:4096]]@u | K1[...]@u]   M=4096 N=32 K=8192 (dense-K streams)
  gemm_f32_wmma<32><<<4096 / 64, blk, 0, stream>>>(K0, n_id, 8192, u, h1, 64, 0,
                                                   8192, nullptr, 1.0f, nullptr, nullptr, 0);
  gemm_f32_wmma<32><<<4096 / 64, blk, 0, stream>>>(K1, n_id, 8192, u, h1, 64, 32,
                                                   8192, nullptr, 1.0f, nullptr, nullptr, 0);
  // 4) u2 scatter of h1
  hipMemsetAsync(u2, 0, (size_t)8192 * 64 * 4, stream);
  scatter_kernel<<<(4096 * 64 + 255) / 256, blk, 0, stream>>>(h1, n_id, u2, 4096, 64, 0);
  // 5) hop2 -> z cols 64..191     M=2048 N=64 K=8192
  gemm_f32_wmma<64><<<2048 / 64, blk, 0, stream>>>(K0, n_id, 8192, u2, z, 192, 64,
                                                   8192, nullptr, 1.0f, nullptr, nullptr, 0);
  gemm_f32_wmma<64><<<2048 / 64, blk, 0, stream>>>(K1, n_id, 8192, u2, z, 192, 128,
                                                   8192, nullptr, 1.0f, nullptr, nullptr, 0);
  // 6) z cols 0..63 = h1[:2048]
  copy_h1_to_z<<<(2048 * 64 + 255) / 256, blk, 0, stream>>>(h1, z);
  // 7) segment-mean + conv input concat
  agg_build_azw<<<(1024 * 192 + 255) / 256, blk, 0, stream>>>(z, edge_ix, azw);
  // 8) stacked weights
  build_wcat<<<(384 * 128 + 255) / 256, blk, 0, stream>>>(W_self, W_neigh, wc);
  // 9) head: hc = azw@Wcat + b_conv ; hm = relu(hc@W1+b1) ; out = hm@W2+b2
  gemm_f32_wmma<128><<<1024 / 64, blk, 0, stream>>>(azw, nullptr, 384, wc, hc, 128, 0,
                                                    384, nullptr, 1.0f, nullptr, b_conv, 0);
  gemm_f32_wmma<64><<<1024 / 64, blk, 0, stream>>>(hc, nullptr, 128, W1, hm, 64, 0,
                                                   128, nullptr, 1.0f, nullptr, b1, 1);
  gemm_f32_wmma<32><<<1024 / 64, blk, 0, stream>>>(hm, nullptr, 64, W2, out, 32, 0,
                                                   64, nullptr, 1.0f, nullptr, b2, 0);
}